// CGIterator_26414048870870
// MI455X (gfx1250) — compile-verified
//
#include <hip/hip_runtime.h>

// ---------------------------------------------------------------------------
// CG tensor-product iterator for MI455X (gfx1250), all GEMMs on the WMMA pipe.
//   feats  : [4, 4096, 7, 128] f32
//   U      : [34, 7, 7, 7]     f32
//   mixerW : [3, 4, 128, 128]  f32
//   iterW  : [2, 4, 128, 128]  f32
//   out    : [4, 4096, 7, 128] f32
//
// WMMA-blocked LDS layouts: for each (K-quad, 16-wide tile) the 64 values a
// wave needs are contiguous in [hi][ln][k&1] order -> one ds_load_b64 per
// operand at (base + lane*8), touching all 64 banks exactly once.
// ---------------------------------------------------------------------------

typedef __attribute__((ext_vector_type(2))) float v2f;
typedef __attribute__((ext_vector_type(8))) float v8f;

#define NDEG  4
#define NSAMP 4096
#define MD    7
#define KCH   128
#define ROWS_PER_DEG (NSAMP * MD)          // 28672
#define UROWS 32                            // (L, c) padded rows: L*8 + c
#define UCOLS 256                           // (l1,a) x (l2,b) : 16*16
#define UBLK  (64 * 2 * 64)                 // blocked Ubig^T: 8192 floats, 32 KB

// fp32 WMMA: D(16x16) = A(16x4) * B(4x16) + C, exact fp32 accumulate.
__device__ __forceinline__ v8f wmma_f32(v2f a, v2f b, v8f c) {
    return __builtin_amdgcn_wmma_f32_16x16x4_f32(
        /*neg_a=*/false, a, /*neg_b=*/false, b,
        /*c_mod=*/(short)0, c, /*reuse_a=*/false, /*reuse_b=*/false);
}

// ---------------------------------------------------------------------------
// Build blocked Ubig^T from the 34 CG paths. One block.
// Logical matrix: UbigT[row = L*8 + c][r = (l1^2+a)*16 + (l2^2+b)], 32 x 256.
// Blocked slot for (row, r):
//   ((r>>2)*2 + (row>>4))*64 + ((r&3)>>1)*32 + (row&15)*2 + (r&1)
// ---------------------------------------------------------------------------
__global__ __launch_bounds__(256) void build_ubig(const float* __restrict__ U,
                                                  float* __restrict__ Ubig) {
    static const signed char pl1[34] = {0,1,2,3, 0,1,1,1,2,2,2,3,3,
                                        0,1,1,1,2,2,2,2,3,3,3,
                                        0,1,1,2,2,2,3,3,3,3};
    static const signed char pl2[34] = {0,1,2,3, 1,0,1,2,1,2,3,2,3,
                                        2,1,2,3,0,1,2,3,1,2,3,
                                        3,2,3,1,2,3,0,1,2,3};
    static const signed char pLL[34] = {0,0,0,0, 1,1,1,1,1,1,1,1,1,
                                        2,2,2,2,2,2,2,2,2,2,2,
                                        3,3,3,3,3,3,3,3,3,3};
    const int tid = threadIdx.x;
    for (int i = tid; i < UBLK; i += 256) Ubig[i] = 0.0f;
    __syncthreads();
    for (int idx = tid; idx < 34 * 343; idx += 256) {
        int p   = idx / 343;
        int rem = idx - p * 343;
        int a = rem / 49;
        int b = (rem / 7) % 7;
        int c = rem % 7;
        int l1 = pl1[p], l2 = pl2[p], L = pLL[p];
        if (a < 2 * l1 + 1 && b < 2 * l2 + 1 && c < 2 * L + 1) {
            int row = L * 8 + c;
            int r   = (l1 * l1 + a) * 16 + (l2 * l2 + b);
            int dst = ((r >> 2) * 2 + (row >> 4)) * 64 +
                      ((r & 3) >> 1) * 32 + (row & 15) * 2 + (r & 1);
            Ubig[dst] = U[idx];
        }
    }
}

// ---------------------------------------------------------------------------
// Per-degree channel mix: Out[l] (+)= A[l] @ W[l],  A rows = (n,m), K = 128.
// Block = 256 thr (8 waves), tile = 128 rows x 128 cols; W staged in LDS in
// WMMA-blocked layout; A global loads software-pipelined one K-step ahead.
// ---------------------------------------------------------------------------
__global__ __launch_bounds__(256) void linear_kernel(const float* __restrict__ A,
                                                     const float* __restrict__ Wt,
                                                     float* __restrict__ Out,
                                                     int residual) {
    __shared__ float sW[KCH * KCH];  // 64 KB, blocked
    const int l   = blockIdx.y;
    const int rb  = blockIdx.x;
    const int tid = threadIdx.x;

    // stage W[l] (row-major [k][j]) into blocked LDS layout
    {
        const float4* Wg = reinterpret_cast<const float4*>(Wt + (size_t)l * KCH * KCH);
#pragma unroll
        for (int i = 0; i < 16; ++i) {
            const int idx4 = tid + i * 256;          // float4 index
            const float4 v = Wg[idx4];
            const int k   = idx4 >> 5;               // 32 float4 per k-row
            const int j0  = (idx4 & 31) * 4;
            const int kq  = (k >> 2) * 8;
            const int hi2 = ((k & 3) >> 1) * 32;
            const int kb  = k & 1;
#pragma unroll
            for (int e = 0; e < 4; ++e) {
                const int j = j0 + e;
                sW[(kq + (j >> 4)) * 64 + hi2 + (j & 15) * 2 + kb] = (&v.x)[e];
            }
        }
    }
    __syncthreads();

    const int w    = tid >> 5;
    const int lane = tid & 31;
    const int ln   = lane & 15;
    const int hi   = lane >> 4;

    const size_t rowbase = (size_t)l * ROWS_PER_DEG + (size_t)rb * 128 + (size_t)w * 16;
    const float* Ap = A + rowbase * KCH + (size_t)ln * KCH + 2 * hi;

    v8f acc[8];
#pragma unroll
    for (int ct = 0; ct < 8; ++ct) acc[ct] = {};

    v2f a2 = *reinterpret_cast<const v2f*>(Ap);      // preload K-step 0
#pragma unroll
    for (int kk = 0; kk < KCH; kk += 4) {
        v2f an = a2;
        if (kk + 4 < KCH)                            // prefetch next K-step
            an = *reinterpret_cast<const v2f*>(Ap + kk + 4);
        const float* wb = &sW[(kk >> 2) * 512 + lane * 2];
#pragma unroll
        for (int ct = 0; ct < 8; ++ct)
            acc[ct] = wmma_f32(a2, *reinterpret_cast<const v2f*>(wb + ct * 64), acc[ct]);
        a2 = an;
    }

    float* Ob = Out + rowbase * KCH;
#pragma unroll
    for (int ct = 0; ct < 8; ++ct) {
#pragma unroll
        for (int vg = 0; vg < 8; ++vg) {
            const int m = vg + 8 * hi;                          // D row
            const size_t off = (size_t)m * KCH + ct * 16 + ln;  // along j: coalesced
            float v = acc[ct][vg];
            if (residual) v += Ob[off];
            Ob[off] = v;
        }
    }
}

// ---------------------------------------------------------------------------
// Tensor product as GEMM: Out^T[(L,c), (n,k)] = Ubig^T[(L,c), r] @ P^T[r, (n,k)]
//   P[r=(l1,a)*16+(l2,b)] = f1[l1][n,a,k] * f2[l2][n,b,k]  (built on the fly)
// One 16-wide (n,k) tile per wave; 2 row-tiles x 64 K-steps = 128 WMMAs.
// ---------------------------------------------------------------------------
__global__ __launch_bounds__(256) void tp_kernel(const float* __restrict__ F1,
                                                 const float* __restrict__ F2,
                                                 const float* __restrict__ Ubig,
                                                 float* __restrict__ Out) {
    __shared__ float sU[UBLK];  // 32 KB, blocked
    const int tid = threadIdx.x;
    {
        const float4* Ug  = reinterpret_cast<const float4*>(Ubig);
        float4*       sU4 = reinterpret_cast<float4*>(sU);
#pragma unroll
        for (int i = 0; i < UBLK / 4 / 256; ++i) sU4[tid + i * 256] = Ug[tid + i * 256];
    }
    __syncthreads();

    const int w    = tid >> 5;
    const int lane = tid & 31;
    const int ln   = lane & 15;
    const int hi   = lane >> 4;

    const int ct = blockIdx.x * 8 + w;        // 32768 column tiles total
    const int n  = ct >> 3;
    const int k  = ((ct & 7) << 4) + ln;      // this lane's channel (coalesced)

    // gather the 16 (l,a) values of f1 and f2 for this (n,k)
    static const signed char l_of[16] = {0, 1,1,1, 2,2,2,2,2, 3,3,3,3,3,3,3};
    float f1v[16], f2v[16];
#pragma unroll
    for (int i = 0; i < 16; ++i) {
        const int l = l_of[i];
        const int a = i - l * l;
        const size_t off = ((size_t)(l * NSAMP + n) * MD + a) * KCH + k;
        f1v[i] = F1[off];
        f2v[i] = F2[off];
    }

    v8f acc0 = {};  // rows (L,c) 0..15
    v8f acc1 = {};  // rows (L,c) 16..31

#pragma unroll
    for (int kk = 0; kk < UCOLS; kk += 4) {
        // B operand of P^T: vgpr0 -> r = kk + 2*hi, vgpr1 -> r = kk+1 + 2*hi
        // (kk % 16 in {0,4,8,12} so the f1 index is lane-uniform; f2 needs select)
        const float b0 = f1v[kk >> 4] * (hi ? f2v[(kk + 2) & 15] : f2v[kk & 15]);
        const float b1 = f1v[(kk + 1) >> 4] * (hi ? f2v[(kk + 3) & 15] : f2v[(kk + 1) & 15]);
        v2f b2; b2.x = b0; b2.y = b1;
        const float* ub = &sU[(kk >> 2) * 128 + lane * 2];
        v2f a0 = *reinterpret_cast<const v2f*>(ub);        // row-tile 0
        v2f a1 = *reinterpret_cast<const v2f*>(ub + 64);   // row-tile 1
        acc0 = wmma_f32(a0, b2, acc0);
        acc1 = wmma_f32(a1, b2, acc1);
    }

    // Store: row r -> (L = r>>3, c = r&7); c == 7 columns are padding (vg == 7).
#pragma unroll
    for (int vg = 0; vg < 7; ++vg) {
        {
            const int r = vg + 8 * hi;
            const int L = r >> 3, c = r & 7;
            Out[((size_t)(L * NSAMP + n) * MD + c) * KCH + k] = acc0[vg];
        }
        {
            const int r = 16 + vg + 8 * hi;
            const int L = r >> 3, c = r & 7;
            Out[((size_t)(L * NSAMP + n) * MD + c) * KCH + k] = acc1[vg];
        }
    }
}

// ---------------------------------------------------------------------------
extern "C" void kernel_launch(void* const* d_in, const int* in_sizes, int n_in,
                              void* d_out, int out_size, void* d_ws, size_t ws_size,
                              hipStream_t stream) {
    (void)in_sizes; (void)n_in; (void)out_size; (void)ws_size;
    const float* feats  = (const float*)d_in[0];
    const float* U      = (const float*)d_in[1];
    const float* mixerW = (const float*)d_in[2];
    const float* iterW  = (const float*)d_in[3];
    float* out = (float*)d_out;
    float* ws  = (float*)d_ws;

    const size_t FEL = (size_t)NDEG * NSAMP * MD * KCH;  // 14,680,064 floats
    float* mixed = ws;                 // 58.7 MB
    float* tpb   = ws + FEL;           // 58.7 MB
    float* ubig  = ws + 2 * FEL;       // 32 KB (blocked)

    hipLaunchKernelGGL(build_ubig, dim3(1), dim3(256), 0, stream, U, ubig);

    const dim3 lgrid(ROWS_PER_DEG / 128, NDEG);  // (224, 4)
    const dim3 blk(256);

    // cur = feats @ mixerW[0]
    hipLaunchKernelGGL(linear_kernel, lgrid, blk, 0, stream,
                       feats, mixerW, out, 0);

    for (int t = 0; t < 2; ++t) {
        // mixed = feats @ mixerW[t+1]
        hipLaunchKernelGGL(linear_kernel, lgrid, blk, 0, stream,
                           feats, mixerW + (size_t)(t + 1) * NDEG * KCH * KCH, mixed, 0);
        // tp = TP(cur, mixed)
        hipLaunchKernelGGL(tp_kernel, dim3(32768 / 8), blk, 0, stream,
                           out, mixed, ubig, tpb);
        // cur += tp @ iterW[t]
        hipLaunchKernelGGL(linear_kernel, lgrid, blk, 0, stream,
                           tpb, iterW + (size_t)t * NDEG * KCH * KCH, out, 1);
    }
}